// GNN_51127290692125
// MI455X (gfx1250) — compile-verified
//
#include <hip/hip_runtime.h>

typedef __attribute__((ext_vector_type(2))) float v2f;
typedef __attribute__((ext_vector_type(8))) float v8f;

#define DFEAT 128

__global__ void GNN_zero_kernel(float* __restrict__ p, int n) {
  int i = blockIdx.x * blockDim.x + threadIdx.x;
  if (i < n) p[i] = 0.0f;
}

// One thread per (edge, feature): gather x[src] and atomically accumulate at dst.
__global__ void GNN_scatter_kernel(const float* __restrict__ x,
                                   const long long* __restrict__ src,
                                   const long long* __restrict__ dst,
                                   float* __restrict__ agg,
                                   float* __restrict__ cnt,
                                   int E) {
  unsigned tid = blockIdx.x * blockDim.x + threadIdx.x;
  if (tid >= (unsigned)E * DFEAT) return;
  int e = tid >> 7;          // DFEAT == 128
  int f = tid & 127;
  int s = (int)src[e];
  int d = (int)dst[e];
  atomicAdd(&agg[d * DFEAT + f], x[s * DFEAT + f]);
  if (f == 0) atomicAdd(&cnt[d], 1.0f);
}

// Fused SAGEConv GEMM: out = act( (agg/cnt) @ Wl + bl + feat @ Wr )
// One wave computes a 16x16 output tile via V_WMMA_F32_16X16X4_F32,
// 8 waves per block share a 128x16 LDS slice of Wl and Wr.
__global__ void __launch_bounds__(256)
GNN_sage_gemm_kernel(const float* __restrict__ feat,  // [N,128] root features
                     const float* __restrict__ agg,   // [N,128] neighbor sums
                     const float* __restrict__ cnt,   // [N]
                     const float* __restrict__ Wl,    // [128,ncols]
                     const float* __restrict__ bl,    // [ncols]
                     const float* __restrict__ Wr,    // [128,ncols]
                     float* __restrict__ out,         // [N,ncols]
                     int N, int ncols, int do_relu) {
  __shared__ float sWl[DFEAT * 16];
  __shared__ float sWr[DFEAT * 16];

  const int c0 = blockIdx.y * 16;  // column tile base
  // Cooperative load of the 128x16 weight slices into LDS.
  for (int i = threadIdx.x; i < DFEAT * 16; i += blockDim.x) {
    int k = i >> 4, j = i & 15;
    sWl[i] = Wl[k * ncols + c0 + j];
    sWr[i] = Wr[k * ncols + c0 + j];
  }
  __syncthreads();

  const int wave = threadIdx.x >> 5;
  const int lane = threadIdx.x & 31;
  const int half = lane >> 4;   // ISA layout: upper half-lanes carry K+2..K+3
  const int idx  = lane & 15;   // A: row index; B/C/D: column index

  const int row_base = (blockIdx.x * 8 + wave) * 16;
  if (row_base >= N) return;    // wave-uniform: EXEC stays all-1s for WMMA

  const int arow = row_base + idx;
  const float c_ = cnt[arow];
  const float inv = 1.0f / fmaxf(c_, 1.0f);

  const float* aggp  = agg  + (size_t)arow * DFEAT;
  const float* featp = feat + (size_t)arow * DFEAT;

  v8f acc = {};
#pragma unroll
  for (int k0 = 0; k0 < DFEAT; k0 += 4) {
    const int ka = k0 + 2 * half;
    // A tiles (16x4 f32): VGPR0 = K=ka, VGPR1 = K=ka+1 per lane-half
    float2 t0 = *(const float2*)(aggp + ka);
    float2 t1 = *(const float2*)(featp + ka);
    v2f am, ax, bL, bR;
    am.x = t0.x * inv; am.y = t0.y * inv;   // mean-aggregated neighbors
    ax.x = t1.x;       ax.y = t1.y;         // root features
    // B tiles (4x16 f32): lane = column, VGPRs hold successive K
    bL.x = sWl[ka * 16 + idx];
    bL.y = sWl[(ka + 1) * 16 + idx];
    bR.x = sWr[ka * 16 + idx];
    bR.y = sWr[(ka + 1) * 16 + idx];
    acc = __builtin_amdgcn_wmma_f32_16x16x4_f32(false, am, false, bL,
                                                (short)0, acc, false, false);
    acc = __builtin_amdgcn_wmma_f32_16x16x4_f32(false, ax, false, bR,
                                                (short)0, acc, false, false);
  }

  const int col = c0 + idx;
  const float bias = bl[col];
#pragma unroll
  for (int v = 0; v < 8; ++v) {
    int r = row_base + v + 8 * half;  // C/D layout: VGPR v -> row v + 8*half
    float d = acc[v] + bias;
    if (do_relu) d = fmaxf(d, 0.0f);
    out[(size_t)r * ncols + col] = d;
  }
}

extern "C" void kernel_launch(void* const* d_in, const int* in_sizes, int n_in,
                              void* d_out, int out_size, void* d_ws, size_t ws_size,
                              hipStream_t stream) {
  const float*     x   = (const float*)d_in[0];
  const long long* ei  = (const long long*)d_in[1];
  const float*     W1l = (const float*)d_in[2];
  const float*     b1l = (const float*)d_in[3];
  const float*     W1r = (const float*)d_in[4];
  const float*     W2l = (const float*)d_in[5];
  const float*     b2l = (const float*)d_in[6];
  const float*     W2r = (const float*)d_in[7];
  float* out = (float*)d_out;

  const int N = 50000;
  const int E = in_sizes[1] / 2;  // edge_index is [2, E] int64

  // Workspace layout: agg[N*128] | cnt[N] | h[N*128]
  float* agg = (float*)d_ws;
  float* cnt = agg + (size_t)N * DFEAT;
  float* h   = cnt + N;

  const long long* src = ei;
  const long long* dstp = ei + E;

  const int zn = N * DFEAT + N;                 // agg + cnt contiguous
  const dim3 zg((zn + 255) / 256);
  const unsigned sthreads = (unsigned)E * DFEAT;
  const dim3 sg((sthreads + 255) / 256);
  const dim3 g1((N / 16 + 7) / 8, DFEAT / 16);  // 391 x 8
  const dim3 g2((N / 16 + 7) / 8, 64 / 16);     // 391 x 4

  // ---- Layer 1 ----
  GNN_zero_kernel<<<zg, 256, 0, stream>>>(agg, zn);
  GNN_scatter_kernel<<<sg, 256, 0, stream>>>(x, src, dstp, agg, cnt, E);
  GNN_sage_gemm_kernel<<<g1, 256, 0, stream>>>(x, agg, cnt, W1l, b1l, W1r,
                                               h, N, DFEAT, 1);
  // ---- Layer 2 ----
  GNN_zero_kernel<<<zg, 256, 0, stream>>>(agg, zn);
  GNN_scatter_kernel<<<sg, 256, 0, stream>>>(h, src, dstp, agg, cnt, E);
  GNN_sage_gemm_kernel<<<g2, 256, 0, stream>>>(h, agg, cnt, W2l, b2l, W2r,
                                               out, N, 64, 0);
}